// PixproLoss_6622839571360
// MI455X (gfx1250) — compile-verified
//
#include <hip/hip_runtime.h>

typedef float v2f __attribute__((ext_vector_type(2)));
typedef float v8f __attribute__((ext_vector_type(8)));

#define EPSV   1e-6f
#define C_DIM  256
#define P_DIM  49
#define P_PAD  64
#define A_PAD  (P_PAD * P_DIM + P_PAD)   // 3200: lets p<64,q<64 index safely; rows >=49 zero

// Per-batch kernel: one workgroup (4 wave32) per batch element.
// Dynamic LDS layout:
//   bls : 256x64 fp32, channel-pair interleaved: (c,p) -> (c>>1)*128 + p*2 + (c&1)
//   mls : same, for moment
//   invnb/invnm : 64 fp32 each (1/max(norm,eps))
//   red : 256 fp32 reduction scratch
//   aLds: 3200 int mask tile (2401 real + zero pad)
extern "C" __global__ __launch_bounds__(128)
void pixpro_batch_kernel(const float* __restrict__ base,
                         const float* __restrict__ moment,
                         const int*   __restrict__ Amat,
                         float* __restrict__ ws_num,
                         float* __restrict__ ws_den)
{
    extern __shared__ float smem[];
    float* bls   = smem;                        // 256*64
    float* mls   = bls + C_DIM * P_PAD;         // 256*64
    float* invnb = mls + C_DIM * P_PAD;         // 64
    float* invnm = invnb + P_PAD;               // 64
    float* red   = invnm + P_PAD;               // 256
    int*   aLds  = (int*)(red + 256);           // 3200

    const int tid   = threadIdx.x;
    const int lane  = tid & 31;
    const int wave  = tid >> 5;
    const int batch = blockIdx.x;
    const size_t gOff = (size_t)batch * (C_DIM * P_DIM);
    const size_t aOff = (size_t)batch * (P_DIM * P_DIM);

    // ---- Stage b, m into LDS with channel-pair interleave ----
    for (int idx = tid; idx < C_DIM * P_DIM; idx += 128) {
        int c = idx / P_DIM;
        int p = idx - c * P_DIM;
        int l = (c >> 1) * (2 * P_PAD) + p * 2 + (c & 1);
        bls[l] = base[gOff + idx];
        mls[l] = moment[gOff + idx];
    }
    // zero the padded columns p in [49, 64)
    for (int idx = tid; idx < C_DIM * (P_PAD - P_DIM); idx += 128) {
        int c = idx / (P_PAD - P_DIM);
        int p = P_DIM + (idx - c * (P_PAD - P_DIM));
        int l = (c >> 1) * (2 * P_PAD) + p * 2 + (c & 1);
        bls[l] = 0.0f;
        mls[l] = 0.0f;
    }
    // ---- Stage mask tile (coalesced), zero-pad tail so rows p>=49 read 0 ----
    for (int idx = tid; idx < A_PAD; idx += 128) {
        aLds[idx] = (idx < P_DIM * P_DIM) ? Amat[aOff + idx] : 0;
    }
    __syncthreads();

    // ---- Inverse norms: 2 threads per position, each over 64 channel-pairs ----
    {
        const v2f* bp = (const v2f*)bls;
        const v2f* mp = (const v2f*)mls;
        int p = tid >> 1;          // 0..63
        int h = tid & 1;           // half selector
        float sb = 0.0f, sm = 0.0f;
        for (int pr = h * 64; pr < h * 64 + 64; ++pr) {
            v2f x = bp[pr * P_PAD + p];
            v2f y = mp[pr * P_PAD + p];
            sb += x.x * x.x + x.y * x.y;
            sm += y.x * y.x + y.y * y.y;
        }
        red[tid]       = sb;
        red[128 + tid] = sm;
    }
    __syncthreads();
    if ((tid & 1) == 0) {
        int p = tid >> 1;
        float SB = red[tid] + red[tid + 1];
        float SM = red[128 + tid] + red[128 + tid + 1];
        invnb[p] = 1.0f / fmaxf(__builtin_sqrtf(SB), EPSV);
        invnm[p] = 1.0f / fmaxf(__builtin_sqrtf(SM), EPSV);
    }
    __syncthreads();

    // ---- WMMA: dots = b^T (49x256) * m (256x49), padded to 64x64 ----
    const v2f* bp = (const v2f*)bls;   // pair index: (c>>1)*P_PAD + p
    const v2f* mp = (const v2f*)mls;

    const int Mbase    = wave * 16;        // wave's M tile
    const int lm       = lane & 15;
    const int kOffPair = lane >> 4;        // 0: K=0,1 ; 1: K=2,3 within a K-group of 4

    // C/D layout: VGPR r -> M = r (lanes 0-15) or r+8 (lanes 16-31); N = lane&15
    const int pbase = Mbase + ((lane >= 16) ? 8 : 0);
    float inb[8];
    #pragma unroll
    for (int r = 0; r < 8; ++r) inb[r] = invnb[pbase + r];

    float num = 0.0f;
    float den = 0.0f;

    for (int nt = 0; nt < 4; ++nt) {
        const int Nbase = nt * 16;
        v8f acc = {};
        #pragma unroll 4
        for (int k = 0; k < C_DIM / 4; ++k) {
            int pairIdx = k * 2 + kOffPair;
            // A fragment: A[M][K] = b[K][M]  (b transposed)
            v2f afrag = bp[pairIdx * P_PAD + Mbase + lm];
            // B fragment: B[K][N] = m[K][N]
            v2f bfrag = mp[pairIdx * P_PAD + Nbase + lm];
            acc = __builtin_amdgcn_wmma_f32_16x16x4_f32(
                false, afrag, false, bfrag, (short)0, acc, false, false);
        }
        const int   q   = Nbase + lm;                         // < 64
        const float qw  = (q < P_DIM) ? 1.0f : 0.0f;          // col guard
        const float iq  = invnm[q];                           // garbage if q>=49, killed by sel
        #pragma unroll
        for (int r = 0; r < 8; ++r) {
            int p = pbase + r;                                // < 64
            int a = aLds[p * P_DIM + q];                      // rows >=49 -> zero pad
            float sel = (a != 0) ? qw : 0.0f;                 // branchless mask
            num += sel * acc[r] * inb[r] * iq;
            den += sel;
        }
    }

    // ---- Block reduction (deterministic tree) ----
    __syncthreads();   // red[] reused; ensure norm phase reads are done
    red[tid]       = num;
    red[128 + tid] = den;
    __syncthreads();
    for (int s = 64; s > 0; s >>= 1) {
        if (tid < s) {
            red[tid]       += red[tid + s];
            red[128 + tid] += red[128 + tid + s];
        }
        __syncthreads();
    }
    if (tid == 0) {
        ws_num[batch] = red[0];
        ws_den[batch] = red[128];
    }
}

// Final deterministic reduction over batch partials -> single scalar.
extern "C" __global__ __launch_bounds__(256)
void pixpro_final_kernel(const float* __restrict__ ws_num,
                         const float* __restrict__ ws_den,
                         float* __restrict__ out, int nbatch)
{
    __shared__ float s[512];
    float n = 0.0f, d = 0.0f;
    for (int i = threadIdx.x; i < nbatch; i += 256) {
        n += ws_num[i];
        d += ws_den[i];
    }
    s[threadIdx.x]       = n;
    s[256 + threadIdx.x] = d;
    __syncthreads();
    for (int st = 128; st > 0; st >>= 1) {
        if (threadIdx.x < st) {
            s[threadIdx.x]       += s[threadIdx.x + st];
            s[256 + threadIdx.x] += s[256 + threadIdx.x + st];
        }
        __syncthreads();
    }
    if (threadIdx.x == 0) {
        out[0] = -(s[0] / s[256]);
    }
}

extern "C" void kernel_launch(void* const* d_in, const int* in_sizes, int n_in,
                              void* d_out, int out_size, void* d_ws, size_t ws_size,
                              hipStream_t stream)
{
    const float* base   = (const float*)d_in[0];
    const float* moment = (const float*)d_in[1];
    const int*   Amat   = (const int*)d_in[2];
    float* out = (float*)d_out;

    const int nbatch = in_sizes[0] / (C_DIM * P_DIM);   // 2048

    float* wsn = (float*)d_ws;
    float* wsd = wsn + nbatch;

    const size_t ldsBytes =
        (size_t)(2 * C_DIM * P_PAD + 2 * P_PAD + 256 + A_PAD) * sizeof(float);

    pixpro_batch_kernel<<<nbatch, 128, ldsBytes, stream>>>(base, moment, Amat, wsn, wsd);
    pixpro_final_kernel<<<1, 256, 0, stream>>>(wsn, wsd, out, nbatch);
}